// GCNLayer_45140106281500
// MI455X (gfx1250) — compile-verified
//
#include <hip/hip_runtime.h>
#include <hip/hip_bf16.h>

// -------- CDNA5 (gfx1250) GCN layer: scatter-sum -> fp32 WMMA GEMM -> fused BN --------
//
// Roofline: GEMM is 1.64 GFLOP over ~77 MB of traffic -> memory bound, so we keep
// full fp32 precision and use V_WMMA_F32_16X16X4_F32 (wave32 matrix pipe) instead
// of down-converting to bf16. The edge scatter (102.4M f32 atomics) dominates; h
// and agg both fit in the 192MB L2 so atomics stay on-chip.

typedef __attribute__((ext_vector_type(2))) float v2f;
typedef __attribute__((ext_vector_type(8))) float v8f;

#define IN_DIM   64
#define HID_DIM  128
#define BN_EPS   1e-5f

// ---------------- zero workspace (agg + column stats) ----------------
__global__ void gcn_zero_kernel(float* __restrict__ p, long n) {
    long i = (long)blockIdx.x * blockDim.x + threadIdx.x;
    if (i < n) p[i] = 0.0f;
}

// ---------------- edge scatter: agg[dst] += h[src] ----------------
// One wave32 per edge: lane L loads h[src][2L..2L+1] (coalesced 256B row) and
// does two global_atomic_add_f32 into agg[dst].
__global__ void gcn_scatter_kernel(const float* __restrict__ h,
                                   const int* __restrict__ src,
                                   const int* __restrict__ dst,
                                   float* __restrict__ agg,
                                   int n_edges) {
    int t    = blockIdx.x * blockDim.x + threadIdx.x;
    int e    = t >> 5;          // 32 lanes per edge
    int lane = t & 31;
    if (e >= n_edges) return;
    int s = src[e];
    int d = dst[e];
    const float2 val = ((const float2*)(h + (long)s * IN_DIM))[lane];
    float* ap = agg + (long)d * IN_DIM + 2 * lane;
    atomicAdd(ap + 0, val.x);
    atomicAdd(ap + 1, val.y);
}

// ---------------- GEMM z = agg @ W + b  (V_WMMA_F32_16X16X4_F32) ----------------
// Block = 256 threads = 8 waves. Block computes a 16-row x 128-col tile of z;
// wave w owns the 16-col tile starting at w*16. K=64 -> 16 chained k=4 WMMAs.
// Epilogue: add bias, store z tile, and fold per-column partial sum / sum-of-
// squares (for BatchNorm) into global accumulators with 2 atomics per column.
__global__ void gcn_gemm_wmma_kernel(const float* __restrict__ agg,
                                     const float* __restrict__ W,   // [64][128]
                                     const float* __restrict__ bias,
                                     float* __restrict__ z,         // [n_rows][128]
                                     float* __restrict__ colsum,    // [128]
                                     float* __restrict__ colsq) {   // [128]
    const int lane = threadIdx.x & 31;
    const int wave = threadIdx.x >> 5;     // 0..7 -> N tile
    const int half = lane >> 4;            // 0 or 1 (lane group)
    const int l15  = lane & 15;
    const int r0   = blockIdx.x * 16;      // row tile base
    const int col  = wave * 16 + l15;      // this lane's output column

    // A fragment base: lane (half,l15) holds A[r0+l15][4k+2*half + {0,1}]
    const float* Arow = agg + (long)(r0 + l15) * IN_DIM + 2 * half;

    v8f acc = {};
#pragma unroll
    for (int k = 0; k < 16; ++k) {
        v2f a;
        a.x = Arow[4 * k + 0];
        a.y = Arow[4 * k + 1];
        // B fragment: lane holds W[4k+2*half + {0,1}][col]
        v2f bf;
        bf.x = W[(long)(4 * k + 2 * half + 0) * HID_DIM + col];
        bf.y = W[(long)(4 * k + 2 * half + 1) * HID_DIM + col];
        // (neg_a, A, neg_b, B, c_mod, C, reuse_a, reuse_b)
        acc = __builtin_amdgcn_wmma_f32_16x16x4_f32(false, a, false, bf,
                                                    (short)0, acc, false, false);
    }

    // Epilogue: bias + store + per-column partial BN stats.
    const float bv = bias[col];
    float s = 0.0f, sq = 0.0f;
#pragma unroll
    for (int v = 0; v < 8; ++v) {
        const int m = v + 8 * half;            // row within tile (C/D layout)
        const float zv = acc[v] + bv;
        z[(long)(r0 + m) * HID_DIM + col] = zv;
        s  += zv;
        sq += zv * zv;
    }
    // lanes L and L+16 hold the same column (different row halves): fold them.
    s  += __shfl_xor(s,  16, 32);
    sq += __shfl_xor(sq, 16, 32);
    if (half == 0) {
        atomicAdd(&colsum[col], s);
        atomicAdd(&colsq[col],  sq);
    }
}

// ---------------- BatchNorm apply (in place over d_out) ----------------
// mean = colsum/N ; var = colsq/N - mean^2 (biased, matches reference)
__global__ void gcn_bn_kernel(float* __restrict__ zio,            // [n_rows][128], in/out
                              const float* __restrict__ colsum,
                              const float* __restrict__ colsq,
                              const float* __restrict__ gamma,
                              const float* __restrict__ beta,
                              int total4, float invN) {
    int i = blockIdx.x * blockDim.x + threadIdx.x;   // over float4 elements
    if (i >= total4) return;
    float4 v = ((const float4*)zio)[i];
    const int c0 = (i & (HID_DIM / 4 - 1)) << 2;
    float r[4] = {v.x, v.y, v.z, v.w};
    float o[4];
#pragma unroll
    for (int j = 0; j < 4; ++j) {
        const int c = c0 + j;
        const float mean = colsum[c] * invN;
        const float var  = colsq[c] * invN - mean * mean;
        o[j] = (r[j] - mean) * rsqrtf(var + BN_EPS) * gamma[c] + beta[c];
    }
    float4 ov = {o[0], o[1], o[2], o[3]};
    ((float4*)zio)[i] = ov;
}

extern "C" void kernel_launch(void* const* d_in, const int* in_sizes, int n_in,
                              void* d_out, int out_size, void* d_ws, size_t ws_size,
                              hipStream_t stream) {
    const float* h     = (const float*)d_in[0];
    const int*   src   = (const int*)  d_in[1];
    const int*   dst   = (const int*)  d_in[2];
    const float* W     = (const float*)d_in[3];
    const float* b     = (const float*)d_in[4];
    const float* gamma = (const float*)d_in[5];
    const float* beta  = (const float*)d_in[6];

    const int n_nodes = in_sizes[0] / IN_DIM;   // 100000
    const int n_edges = in_sizes[1];            // 1600000

    // Workspace layout: [agg: n_nodes*64][colsum: 128][colsq: 128]
    float* agg    = (float*)d_ws;
    float* colsum = agg + (long)n_nodes * IN_DIM;
    float* colsq  = colsum + HID_DIM;
    float* z      = (float*)d_out;              // stage z directly in d_out

    // 1) zero agg + stats (contiguous)
    const long nzero = (long)n_nodes * IN_DIM + 2 * HID_DIM;
    gcn_zero_kernel<<<(int)((nzero + 255) / 256), 256, 0, stream>>>(agg, nzero);

    // 2) edge scatter (one wave32 per edge)
    const int scatter_threads_blocks = (int)(((long)n_edges * 32 + 255) / 256);
    gcn_scatter_kernel<<<scatter_threads_blocks, 256, 0, stream>>>(h, src, dst, agg, n_edges);

    // 3) fp32 WMMA GEMM + fused BN partial stats (n_nodes divisible by 16)
    gcn_gemm_wmma_kernel<<<n_nodes / 16, 256, 0, stream>>>(agg, W, b, z, colsum, colsq);

    // 4) BN apply in place over d_out
    const int total4 = n_nodes * (HID_DIM / 4);
    gcn_bn_kernel<<<(total4 + 255) / 256, 256, 0, stream>>>(z, colsum, colsq, gamma, beta,
                                                            total4, 1.0f / (float)n_nodes);
}